// GCN_34746285424763
// MI455X (gfx1250) — compile-verified
//
#include <hip/hip_runtime.h>

// GCN forward: out = spmm(relu(spmm(x@W1)+b1) @ W2) + b2
// MI455X strategy: SpMM via L2-resident fp32 atomics (H0/agg/H2 all fit in
// 192MB L2), GEMMs via V_WMMA_F32_16X16X4_F32 (full fp32 precision; GEMM cost
// is negligible vs ~100us of SpMM traffic at 23.3 TB/s).

#define N_NODES 100000
#define N_EDGES 1600000
#define NFEAT   256
#define NHID    128
#define NCLASS  40

typedef __attribute__((ext_vector_type(2))) float v2f;
typedef __attribute__((ext_vector_type(8))) float v8f;

// ---------------- GEMM1: H0[100000,128] = x[100000,256] @ W1[256,128] -------
// Block = 256 threads = 8 waves; block mt handles M-tile mt (16 rows),
// wave w handles N-tile w (16 cols). K loop in steps of 4 (fp32 WMMA).
__global__ __launch_bounds__(256)
void gcn_gemm1(const float* __restrict__ x, const float* __restrict__ W1,
               float* __restrict__ H0) {
  const int mt   = blockIdx.x;            // 0..6249
  const int wave = threadIdx.x >> 5;      // 0..7 -> N tile
  const int lane = threadIdx.x & 31;
  const int half = lane >> 4;             // 0: K%4 in {0,1}, 1: {2,3}
  const int mn   = lane & 15;             // A row / B col within tile
  const int m0 = mt * 16;
  const int n0 = wave * 16;

  const float* arow = x  + (size_t)(m0 + mn) * NFEAT;
  const float* bcol = W1 + n0 + mn;

  v8f c = {};
  #pragma unroll 4
  for (int k = 0; k < NFEAT; k += 4) {
    const int k0 = k + 2 * half;
    v2f a, b;
    a.x = arow[k0];
    a.y = arow[k0 + 1];
    b.x = bcol[(size_t)k0 * NHID];
    b.y = bcol[(size_t)(k0 + 1) * NHID];
    c = __builtin_amdgcn_wmma_f32_16x16x4_f32(
        /*neg_a=*/false, a, /*neg_b=*/false, b,
        /*c_mod=*/(short)0, c, /*reuse_a=*/false, /*reuse_b=*/false);
  }
  // C/D layout: VGPR j holds (M = j + 8*half, N = mn)
  float* crow = H0 + (size_t)m0 * NHID + n0 + mn;
  #pragma unroll
  for (int j = 0; j < 8; ++j)
    crow[(size_t)(j + 8 * half) * NHID] = c[j];
}

// ---------------- SpMM1: agg[dst] += w * H0[src]  (wave per edge) ----------
__global__ __launch_bounds__(256)
void gcn_spmm1(const int* __restrict__ esrc, const int* __restrict__ edst,
               const float* __restrict__ ew, const float* __restrict__ H0,
               float* __restrict__ agg) {
  const long long tid = (long long)blockIdx.x * blockDim.x + threadIdx.x;
  const int e = (int)(tid >> 5);
  if (e >= N_EDGES) return;
  const int lane = threadIdx.x & 31;
  const int s = esrc[e];
  const int d = edst[e];
  const float w = ew[e];
  // 32 lanes x float4 = 128 floats = one row; coalesced 512B gather.
  const float4 hv = *(const float4*)(H0 + (size_t)s * NHID + lane * 4);
  float* o = agg + (size_t)d * NHID + lane * 4;
  atomicAdd(o + 0, w * hv.x);
  atomicAdd(o + 1, w * hv.y);
  atomicAdd(o + 2, w * hv.z);
  atomicAdd(o + 3, w * hv.w);
}

// ---- GEMM2 fused: H2[100000,40] = relu(agg + b1) @ W2[128,40] --------------
// 3 waves/block, N padded to 48 with guards at col<40.
__global__ __launch_bounds__(96)
void gcn_gemm2(const float* __restrict__ agg, const float* __restrict__ b1,
               const float* __restrict__ W2, float* __restrict__ H2) {
  const int mt   = blockIdx.x;            // 0..6249
  const int wave = threadIdx.x >> 5;      // 0..2 -> N tile
  const int lane = threadIdx.x & 31;
  const int half = lane >> 4;
  const int mn   = lane & 15;
  const int m0  = mt * 16;
  const int col = wave * 16 + mn;
  const bool colok = (col < NCLASS);

  const float* arow = agg + (size_t)(m0 + mn) * NHID;

  v8f c = {};
  #pragma unroll 4
  for (int k = 0; k < NHID; k += 4) {
    const int k0 = k + 2 * half;
    v2f a, b;
    a.x = fmaxf(arow[k0]     + b1[k0],     0.0f);   // fused bias + ReLU
    a.y = fmaxf(arow[k0 + 1] + b1[k0 + 1], 0.0f);
    b.x = colok ? W2[(size_t)k0 * NCLASS + col]       : 0.0f;
    b.y = colok ? W2[(size_t)(k0 + 1) * NCLASS + col] : 0.0f;
    c = __builtin_amdgcn_wmma_f32_16x16x4_f32(
        false, a, false, b, (short)0, c, false, false);
  }
  if (colok) {
    #pragma unroll
    for (int j = 0; j < 8; ++j)
      H2[(size_t)(m0 + j + 8 * half) * NCLASS + col] = c[j];
  }
}

// ---------------- SpMM2: out[dst] += w * H2[src]  (thread per edge,class) ---
__global__ __launch_bounds__(256)
void gcn_spmm2(const int* __restrict__ esrc, const int* __restrict__ edst,
               const float* __restrict__ ew, const float* __restrict__ H2,
               float* __restrict__ out) {
  const long long tid = (long long)blockIdx.x * blockDim.x + threadIdx.x;
  if (tid >= (long long)N_EDGES * NCLASS) return;
  const int e = (int)(tid / NCLASS);
  const int cc = (int)(tid % NCLASS);
  const int s = esrc[e];
  const int d = edst[e];
  atomicAdd(&out[(size_t)d * NCLASS + cc], ew[e] * H2[(size_t)s * NCLASS + cc]);
}

// ---------------- init kernels ---------------------------------------------
__global__ __launch_bounds__(256)
void gcn_zero4(float4* __restrict__ p, long long n4) {
  const long long i = (long long)blockIdx.x * blockDim.x + threadIdx.x;
  if (i < n4) p[i] = make_float4(0.f, 0.f, 0.f, 0.f);
}

__global__ __launch_bounds__(256)
void gcn_init_out(float* __restrict__ out, const float* __restrict__ b2) {
  const long long i = (long long)blockIdx.x * blockDim.x + threadIdx.x;
  if (i < (long long)N_NODES * NCLASS) out[i] = b2[i % NCLASS];
}

extern "C" void kernel_launch(void* const* d_in, const int* in_sizes, int n_in,
                              void* d_out, int out_size, void* d_ws, size_t ws_size,
                              hipStream_t stream) {
  const float* x    = (const float*)d_in[0];
  const int*   esrc = (const int*)  d_in[1];
  const int*   edst = (const int*)  d_in[2];
  const float* ew   = (const float*)d_in[3];
  const float* W1   = (const float*)d_in[4];
  const float* b1   = (const float*)d_in[5];
  const float* W2   = (const float*)d_in[6];
  const float* b2   = (const float*)d_in[7];
  float* out = (float*)d_out;

  // Workspace layout: [H0 | agg]; H2 reuses H0's region once H0 is dead.
  float* H0  = (float*)d_ws;                                    // 51.2 MB
  float* agg = H0 + (size_t)N_NODES * NHID;                     // 51.2 MB
  float* H2  = H0;                                              // reuse

  // 1) agg = 0
  {
    const long long n4 = (long long)N_NODES * NHID / 4;
    gcn_zero4<<<(unsigned)((n4 + 255) / 256), 256, 0, stream>>>((float4*)agg, n4);
  }
  // 2) H0 = x @ W1
  gcn_gemm1<<<N_NODES / 16, 256, 0, stream>>>(x, W1, H0);
  // 3) agg += spmm
  {
    const long long nt = (long long)N_EDGES * 32;
    gcn_spmm1<<<(unsigned)((nt + 255) / 256), 256, 0, stream>>>(esrc, edst, ew, H0, agg);
  }
  // 4) H2 = relu(agg + b1) @ W2
  gcn_gemm2<<<N_NODES / 16, 96, 0, stream>>>(agg, b1, W2, H2);
  // 5) out = b2 (broadcast)
  {
    const long long n = (long long)N_NODES * NCLASS;
    gcn_init_out<<<(unsigned)((n + 255) / 256), 256, 0, stream>>>(out, b2);
  }
  // 6) out += spmm
  {
    const long long nt = (long long)N_EDGES * NCLASS;
    gcn_spmm2<<<(unsigned)((nt + 255) / 256), 256, 0, stream>>>(esrc, edst, ew, H2, out);
  }
}